// ECSTFL_13056700580199
// MI455X (gfx1250) — compile-verified
//
#include <hip/hip_runtime.h>
#include <cstdint>

#define N_PTS   8192
#define D_FEAT  64
#define NCLS    100
#define EPSV    1e-6f

#define THREADS 256
#define ROWBLKS (N_PTS / THREADS)            // 32 (also used by helper kernels)
#define TI      256                          // rows per block (1 per lane, 8 waves)
#define JCOLS   256                          // columns per block (the J tile)
#define TJ      64                           // columns per LDS sub-tile
#define TILES   (JCOLS / TJ)                 // 4
#define NTILE   (N_PTS / TI)                 // 32
#define NBLOCKS (NTILE * (NTILE + 1) / 2)    // 528 triangular tile-pairs
#define CHUNKS  ((TJ * D_FEAT * 4) / 16 / THREADS)  // 4 async b128 per thread per stage

typedef __attribute__((ext_vector_type(2))) float v2f;
typedef __attribute__((ext_vector_type(8))) float v8f;

__global__ __launch_bounds__(THREADS) void ec_zero(float* posSum, float* invSum,
                                                   int* hist, float* out) {
  int i = blockIdx.x * THREADS + threadIdx.x;
  if (i < N_PTS) { posSum[i] = 0.f; invSum[i] = 0.f; }
  if (i < NCLS)  hist[i] = 0;
  if (i == 0)    out[0] = 0.f;
}

__global__ __launch_bounds__(THREADS) void ec_hist(const int* __restrict__ labels,
                                                   int* __restrict__ hist) {
  int i = blockIdx.x * THREADS + threadIdx.x;
  atomicAdd(&hist[labels[i]], 1);
}

// Symmetric tile-pair kernel: block handles rows of tile I (registers) vs
// columns of tile J (LDS, async-staged).  For I<J every pair is computed once
// and credited to both row i (registers) and column j (cross-lane reduce ->
// LDS accumulators -> one global atomic per column).
__global__ __launch_bounds__(THREADS) void ec_main(const float* __restrict__ feat,
                                                   const int* __restrict__ labels,
                                                   float* __restrict__ posSum,
                                                   float* __restrict__ invSum) {
  __shared__ float sTile[2][TJ * D_FEAT];   // 2 x 16 KB double buffer
  __shared__ int   sLab[JCOLS];             // column labels
  __shared__ float jPos[JCOLS];             // j-side accumulators
  __shared__ float jInv[JCOLS];

  const int tid = threadIdx.x;

  // decode triangular block index -> (I, J) with I <= J
  int b = blockIdx.x, I = 0;
  while (b >= NTILE - I) { b -= NTILE - I; ++I; }
  const int J = I + b;
  const bool offdiag = (I != J);

  const int row   = I * TI + tid;
  const int jBase = J * JCOLS;

  for (int k = tid; k < JCOLS; k += THREADS) {
    sLab[k] = labels[jBase + k];
    jPos[k] = 0.f;
    jInv[k] = 0.f;
  }

  // this lane's row lives in registers for the whole kernel (64 VGPRs)
  float4 r[16];
  const float4* fp = (const float4*)(feat + (size_t)row * D_FEAT);
  #pragma unroll
  for (int c = 0; c < 16; ++c) r[c] = fp[c];
  const int myLab = labels[row];

  // CDNA5 async copy: contiguous 16KB column sub-tile -> LDS (ASYNCcnt-tracked)
  auto stage = [&](int buf, int t) {
    const char* g0 = (const char*)feat + ((size_t)(jBase + t * TJ)) * (D_FEAT * 4);
    unsigned l0 = (unsigned)(uintptr_t)&sTile[buf][0];
    #pragma unroll
    for (int k = 0; k < CHUNKS; ++k) {
      unsigned off = (unsigned)(tid + k * THREADS) * 16u;
      unsigned lds = l0 + off;
      const char* g = g0 + off;
      asm volatile("global_load_async_to_lds_b128 %0, %1, off"
                   :: "v"(lds), "v"(g) : "memory");
    }
  };

  stage(0, 0);

  float posAcc = 0.f, negAcc = 0.f;

  for (int t = 0; t < TILES; ++t) {
    const int buf = t & 1;
    if (t + 1 < TILES) {
      stage(buf ^ 1, t + 1);                               // prefetch next sub-tile
      asm volatile("s_wait_asynccnt %0" :: "i"(CHUNKS) : "memory"); // oldest batch done
    } else {
      asm volatile("s_wait_asynccnt 0" ::: "memory");
    }
    __syncthreads();                                       // sub-tile visible to all waves

    const float* base = &sTile[buf][0];
    #pragma unroll 2
    for (int jj = 0; jj < TJ; ++jj) {
      const float4* cp = (const float4*)(base + jj * D_FEAT);
      float s0 = 0.f, s1 = 0.f, s2 = 0.f, s3 = 0.f;
      #pragma unroll
      for (int c = 0; c < 16; c += 4) {
        float4 v0 = cp[c], v1 = cp[c + 1], v2 = cp[c + 2], v3 = cp[c + 3];
        s0 += __builtin_fabsf(r[c].x - v0.x) + __builtin_fabsf(r[c].y - v0.y) +
              __builtin_fabsf(r[c].z - v0.z) + __builtin_fabsf(r[c].w - v0.w);
        s1 += __builtin_fabsf(r[c + 1].x - v1.x) + __builtin_fabsf(r[c + 1].y - v1.y) +
              __builtin_fabsf(r[c + 1].z - v1.z) + __builtin_fabsf(r[c + 1].w - v1.w);
        s2 += __builtin_fabsf(r[c + 2].x - v2.x) + __builtin_fabsf(r[c + 2].y - v2.y) +
              __builtin_fabsf(r[c + 2].z - v2.z) + __builtin_fabsf(r[c + 2].w - v2.w);
        s3 += __builtin_fabsf(r[c + 3].x - v3.x) + __builtin_fabsf(r[c + 3].y - v3.y) +
              __builtin_fabsf(r[c + 3].z - v3.z) + __builtin_fabsf(r[c + 3].w - v3.w);
      }
      float dist = (s0 + s1) + (s2 + s3);
      const int jAbs = t * TJ + jj;
      const int lj   = sLab[jAbs];
      float inv = __builtin_amdgcn_rcpf(dist + EPSV);      // v_rcp_f32
      bool p = (lj == myLab);
      float pc = p ? dist : 0.f;                           // diagonal pair: dist==0
      float ic = p ? 0.f  : inv;
      posAcc += pc;
      negAcc += ic;
      if (offdiag) {                                       // uniform branch
        // j-side credit: wave-reduce then LDS atomic (DS pipe, hides under VALU)
        #pragma unroll
        for (int m = 16; m >= 1; m >>= 1) {
          pc += __shfl_xor(pc, m, 32);
          ic += __shfl_xor(ic, m, 32);
        }
        if ((tid & 31) == 0) {
          atomicAdd(&jPos[jAbs], pc);
          atomicAdd(&jInv[jAbs], ic);
        }
      }
    }
    __syncthreads();   // all waves done with buf before it is re-staged
  }

  atomicAdd(&posSum[row], posAcc);
  atomicAdd(&invSum[row], negAcc);
  if (offdiag) {
    __syncthreads();   // all LDS j-side atomics complete
    atomicAdd(&posSum[jBase + tid], jPos[tid]);
    atomicAdd(&invSum[jBase + tid], jInv[tid]);
  }
}

__global__ __launch_bounds__(THREADS) void ec_final(const float* __restrict__ posSum,
                                                    const float* __restrict__ invSum,
                                                    const int* __restrict__ labels,
                                                    const int* __restrict__ hist,
                                                    float* __restrict__ out) {
  int i = blockIdx.x * THREADS + threadIdx.x;
  float cnt        = (float)hist[labels[i]];               // same-class count incl. self
  float attractive = posSum[i] / (cnt + EPSV);             // (pos-1)+1+eps == cnt+eps
  float repulsive  = invSum[i] / ((float)N_PTS - cnt + 1.0f + EPSV);
  float tot        = attractive * repulsive * (1.0f / (float)N_PTS);

  // Exact fp32 cross-lane sum via V_WMMA_F32_16X16X4_F32 with B = ones:
  // D[m][n] = t[m] + t[m+16]; fold each lane's 8 D rows + one xor-16 swap.
  v2f a; a[0] = tot; a[1] = 0.f;
  v2f b; b[0] = 1.f; b[1] = 1.f;
  v8f c = {};
  v8f d = __builtin_amdgcn_wmma_f32_16x16x4_f32(false, a, false, b,
                                                (short)0, c, false, false);
  float s = ((d[0] + d[1]) + (d[2] + d[3])) + ((d[4] + d[5]) + (d[6] + d[7]));
  s += __shfl_xor(s, 16, 32);
  if ((threadIdx.x & 31) == 0)
    atomicAdd(out, s * (1.0f / (float)N_PTS));             // mean over rows
}

extern "C" void kernel_launch(void* const* d_in, const int* in_sizes, int n_in,
                              void* d_out, int out_size, void* d_ws, size_t ws_size,
                              hipStream_t stream) {
  (void)in_sizes; (void)n_in; (void)out_size; (void)ws_size;
  const float* feat   = (const float*)d_in[0];
  const int*   labels = (const int*)d_in[1];
  float* out    = (float*)d_out;
  float* posSum = (float*)d_ws;                 // 8192 f32
  float* invSum = posSum + N_PTS;               // 8192 f32
  int*   hist   = (int*)(invSum + N_PTS);       // 100 i32

  ec_zero <<<ROWBLKS, THREADS, 0, stream>>>(posSum, invSum, hist, out);
  ec_hist <<<ROWBLKS, THREADS, 0, stream>>>(labels, hist);
  ec_main <<<NBLOCKS, THREADS, 0, stream>>>(feat, labels, posSum, invSum);
  ec_final<<<ROWBLKS, THREADS, 0, stream>>>(posSum, invSum, labels, hist, out);
}